// InputTransition_8804682957051
// MI455X (gfx1250) — compile-verified
//
#include <hip/hip_runtime.h>

typedef __attribute__((ext_vector_type(2))) float v2f;
typedef __attribute__((ext_vector_type(8))) float v8f;

#define NVOX 400000
#define KTAPS 125
#define NCH 16
#define EPSBN 1e-5f

// ---------------------------------------------------------------------------
// Kernel 0: zero the 32-float BN accumulator (sums[16] ++ sumsq[16]) in d_ws
// ---------------------------------------------------------------------------
__global__ void __launch_bounds__(32) zero_acc_kernel(float* __restrict__ acc) {
  acc[threadIdx.x] = 0.0f;
}

// ---------------------------------------------------------------------------
// Kernel 1: gather-GEMM via V_WMMA_F32_16X16X4_F32.
//   Each wave computes a 16-voxel x 16-channel tile:
//     D[m,d] = sum_k x[nbr[k, row_m]] * W[k,d]
//   A (16x4 f32): lanes 0-15 hold taps (k0,k0+1), lanes 16-31 hold (k0+2,k0+3)
//   B (4x16 f32): same split, channels across lanes
//   D (16x16 f32): VGPR v -> row M = v + 8*(lane>=16), col = lane&15
//   Also accumulates per-channel sum / sum-of-squares for BatchNorm.
// ---------------------------------------------------------------------------
__global__ void __launch_bounds__(256) conv_wmma_kernel(
    const float* __restrict__ x, const float* __restrict__ W,
    const float* __restrict__ bias, const int* __restrict__ nbr,
    float* __restrict__ out, float* __restrict__ acc) {
  __shared__ float lds_sum[NCH];
  __shared__ float lds_sq[NCH];
  const int tid = threadIdx.x;
  if (tid < NCH) { lds_sum[tid] = 0.f; lds_sq[tid] = 0.f; }
  __syncthreads();

  const int lane = tid & 31;
  const int wave = tid >> 5;
  const int half = lane >> 4;   // 0: K pair (k0,k0+1), 1: (k0+2,k0+3)
  const int sub  = lane & 15;   // A-row (voxel) index / B-col (channel) index
  const int tile = blockIdx.x * 8 + wave;     // 25000 tiles of 16 voxels
  const int row  = tile * 16 + sub;           // voxel id for A rows

  v8f c = {};
  for (int kk = 0; kk < 128; kk += 4) {       // K padded 125 -> 128
    const int k0 = kk + 2 * half;
    const int k1 = k0 + 1;
    float a0 = 0.f, a1 = 0.f, b0 = 0.f, b1 = 0.f;
    if (k0 < KTAPS) { a0 = x[nbr[k0 * NVOX + row]]; b0 = W[k0 * NCH + sub]; }
    if (k1 < KTAPS) { a1 = x[nbr[k1 * NVOX + row]]; b1 = W[k1 * NCH + sub]; }
    v2f av; av.x = a0; av.y = a1;
    v2f bv; bv.x = b0; bv.y = b1;
    // D = A x B + C  (fp32 in, fp32 accumulate; EXEC all-ones here)
    c = __builtin_amdgcn_wmma_f32_16x16x4_f32(
        /*neg_a=*/false, av, /*neg_b=*/false, bv,
        /*c_mod=*/(short)0, c, /*reuse_a=*/false, /*reuse_b=*/false);
  }

  // Bias add, store pre-BN activations, accumulate BN statistics.
  const float bch = bias[sub];
  const long base = ((long)tile * 16 + 8 * half) * NCH + sub;
  float lsum = 0.f, lsq = 0.f;
#pragma unroll
  for (int v = 0; v < 8; ++v) {
    const float val = c[v] + bch;   // element (row = tile*16 + v + 8*half, ch = sub)
    out[base + v * NCH] = val;
    lsum += val;
    lsq  += val * val;
  }
  atomicAdd(&lds_sum[sub], lsum);   // ds_add_f32
  atomicAdd(&lds_sq[sub],  lsq);
  __syncthreads();
  if (tid < NCH)            atomicAdd(&acc[tid], lds_sum[tid]);         // global_atomic_add_f32
  else if (tid < 2 * NCH)   atomicAdd(&acc[tid], lds_sq[tid - NCH]);
}

// ---------------------------------------------------------------------------
// Kernel 2: BatchNorm apply + tiled residual + ELU, in place over d_out.
// float4 vectorized: each thread handles 4 consecutive channels of one voxel.
// ---------------------------------------------------------------------------
__global__ void __launch_bounds__(256) bn_res_elu_kernel(
    const float* __restrict__ x, const float* __restrict__ gamma,
    const float* __restrict__ beta, const float* __restrict__ acc,
    float* __restrict__ out) {
  const long i = (long)blockIdx.x * blockDim.x + threadIdx.x;
  if (i >= (long)NVOX * NCH / 4) return;
  const long e  = i * 4;
  const int  n  = (int)(e >> 4);
  const int  d0 = (int)(e & 15);
  const float xn    = x[n];              // residual: x tiled across all 16 channels
  const float inv_n = 1.0f / (float)NVOX;

  float4 v = ((float4*)out)[i];
  float r[4] = {v.x, v.y, v.z, v.w};
#pragma unroll
  for (int j = 0; j < 4; ++j) {
    const int d = d0 + j;
    const float mean  = acc[d] * inv_n;
    const float var   = acc[NCH + d] * inv_n - mean * mean;  // biased var
    const float scale = gamma[d] * rsqrtf(var + EPSBN);
    float z = (r[j] - mean) * scale + beta[d] + xn;
    r[j] = z > 0.f ? z : (__expf(z) - 1.0f);                 // ELU
  }
  v.x = r[0]; v.y = r[1]; v.z = r[2]; v.w = r[3];
  ((float4*)out)[i] = v;
}

// ---------------------------------------------------------------------------
extern "C" void kernel_launch(void* const* d_in, const int* in_sizes, int n_in,
                              void* d_out, int out_size, void* d_ws, size_t ws_size,
                              hipStream_t stream) {
  const float* x     = (const float*)d_in[0];  // [N,1]
  const float* W     = (const float*)d_in[1];  // [K,1,16]
  const float* bias  = (const float*)d_in[2];  // [16]
  const float* gamma = (const float*)d_in[3];  // [16]
  const float* beta  = (const float*)d_in[4];  // [16]
  const int*   nbr   = (const int*)d_in[5];    // [K,N]
  float* out = (float*)d_out;                  // [N,16]
  float* acc = (float*)d_ws;                   // 32 floats: sums[16] ++ sumsq[16]

  zero_acc_kernel<<<1, 32, 0, stream>>>(acc);

  const int tiles  = NVOX / 16;                // 25000
  const int blocks = tiles / 8;                // 3125 (8 waves per 256-thread block)
  conv_wmma_kernel<<<blocks, 256, 0, stream>>>(x, W, bias, nbr, out, acc);

  const long elems4 = (long)NVOX * NCH / 4;    // 1.6M float4 groups
  const int  grid2  = (int)((elems4 + 255) / 256);
  bn_res_elu_kernel<<<grid2, 256, 0, stream>>>(x, gamma, beta, acc, out);
}